// Volume_Render_47588237639987
// MI455X (gfx1250) — compile-verified
//
#include <hip/hip_runtime.h>
#include <stdint.h>

typedef __attribute__((ext_vector_type(4))) unsigned int v4u;
typedef __attribute__((ext_vector_type(4))) int          v4i;
typedef __attribute__((ext_vector_type(8))) int          v8i;

#define NS   128      // samples per ray
#define RPB  8        // rays per block (one wave32 per ray)
#define EPSV 1e-10f
#define INFV 1e10f

// --- TDM: async DMA of a contiguous run of `nelem` f32 from global -> LDS ---
// Builds D# group0/group1 per CDNA5 ISA (CDNA5 ISA):
//   g0: count=1 | lds_addr | global_addr[56:0] | type=2
//   g1: data_size=4B, tensor_dim0=nelem, tile_dim0=nelem (1-D tile), stride=nelem
__device__ __forceinline__ void tdm_load_contig_f32(const float* gsrc,
                                                    uint32_t lds_byte_addr,
                                                    uint32_t nelem) {
  uint64_t ga = (uint64_t)(uintptr_t)gsrc;
  v4u g0;
  g0.x = 1u;                                               // count=1 (valid), user mode
  g0.y = lds_byte_addr;                                    // lds_addr (bytes)
  g0.z = (uint32_t)ga;                                     // global_addr[31:0]
  g0.w = (uint32_t)((ga >> 32) & 0x01FFFFFFu) | (2u << 30);// global_addr[56:32] | type=2

  v8i g1;
  g1[0] = (int)(2u << 16);                                 // data_size=2 (4 bytes)
  g1[1] = (int)((nelem & 0xFFFFu) << 16);                  // tensor_dim0[15:0]
  g1[2] = (int)(((nelem >> 16) & 0xFFFFu) | (1u << 16));   // tensor_dim0[31:16] | tensor_dim1=1
  g1[3] = (int)((nelem & 0xFFFFu) << 16);                  // tensor_dim1[31:16]=0 | tile_dim0
  g1[4] = 0;                                               // tile_dim1=0 (unused), tile_dim2=0
  g1[5] = (int)nelem;                                      // tensor_dim0_stride[31:0]
  g1[6] = 0;                                               // stride0[47:32] | stride1[15:0]
  g1[7] = 0;                                               // stride1[47:16]

  v4i z4 = {0, 0, 0, 0};
#if defined(__clang_major__) && (__clang_major__ >= 23)
  v8i z8 = {0, 0, 0, 0, 0, 0, 0, 0};
  __builtin_amdgcn_tensor_load_to_lds(g0, g1, z4, z4, z8, 0);
#else
  __builtin_amdgcn_tensor_load_to_lds(g0, g1, z4, z4, 0);
#endif
}

__global__ __launch_bounds__(RPB * 32) void volume_render_kernel(
    const float* __restrict__ sigmas,     // [N, 128]
    const float* __restrict__ rgbs,       // [N, 128, 3]
    const float* __restrict__ t_inters,   // [N, 128, 2]
    float* __restrict__ depths,           // [N]
    float* __restrict__ colors,           // [N, 3]
    float* __restrict__ weights) {        // [N, 128]
  __shared__ __align__(16) float s_sig[RPB * NS];
  __shared__ __align__(16) float s_t  [RPB * NS * 2];
  __shared__ __align__(16) float s_rgb[RPB * NS * 3];

  const int lane   = threadIdx.x & 31;
  const int wave   = threadIdx.x >> 5;
  const int block0 = blockIdx.x * RPB;

  // Wave 0 DMAs this block's 24KB working set into LDS via the Tensor Data
  // Mover, then waits on TENSORcnt; other waves park at the barrier.
  if (wave == 0) {
    tdm_load_contig_f32(sigmas   + (size_t)block0 * NS,     (uint32_t)(uintptr_t)s_sig, RPB * NS);
    tdm_load_contig_f32(t_inters + (size_t)block0 * NS * 2, (uint32_t)(uintptr_t)s_t,   RPB * NS * 2);
    tdm_load_contig_f32(rgbs     + (size_t)block0 * NS * 3, (uint32_t)(uintptr_t)s_rgb, RPB * NS * 3);
    __builtin_amdgcn_s_wait_tensorcnt(0);
  }
  __syncthreads();

  const int ray = block0 + wave;
  const float4* sig4 = (const float4*)(s_sig + wave * NS);
  const float4* t4   = (const float4*)(s_t   + wave * NS * 2);
  const float4* rgb4 = (const float4*)(s_rgb + wave * NS * 3);

  // Lane owns samples [4*lane, 4*lane+3]
  float4 sg = sig4[lane];
  float4 ta = t4[lane * 2 + 0];          // t pairs for samples 4l, 4l+1
  float4 tb = t4[lane * 2 + 1];          // t pairs for samples 4l+2, 4l+3
  float4 r0 = rgb4[lane * 3 + 0];
  float4 r1 = rgb4[lane * 3 + 1];
  float4 r2 = rgb4[lane * 3 + 2];

  // sample centers
  float tc0 = 0.5f * (ta.x + ta.y);
  float tc1 = 0.5f * (ta.z + ta.w);
  float tc2 = 0.5f * (tb.x + tb.y);
  float tc3 = 0.5f * (tb.z + tb.w);

  // neighbor diffs; last overall sample gets INF (finite 1e10 like reference)
  float tcn = __shfl_down(tc0, 1);
  float d0 = tc1 - tc0;
  float d1 = tc2 - tc1;
  float d2 = tc3 - tc2;
  float d3 = (lane == 31) ? INFV : (tcn - tc3);

  // alphas
  float a0 = 1.0f - __expf(-sg.x * d0);
  float a1 = 1.0f - __expf(-sg.y * d1);
  float a2 = 1.0f - __expf(-sg.z * d2);
  float a3 = 1.0f - __expf(-sg.w * d3);

  // exclusive cumprod of g = 1 - a + eps
  float g0v = 1.0f - a0 + EPSV;
  float g1v = 1.0f - a1 + EPSV;
  float g2v = 1.0f - a2 + EPSV;
  float g3v = 1.0f - a3 + EPSV;
  float p1 = g0v;            // in-lane exclusive prefixes (p0 = 1)
  float p2 = p1 * g1v;
  float p3 = p2 * g2v;
  float tot = p3 * g3v;      // product of this lane's 4 factors

  // wave32 inclusive multiplicative scan of `tot`, then shift -> exclusive
  float incl = tot;
#pragma unroll
  for (int off = 1; off < 32; off <<= 1) {
    float y = __shfl_up(incl, off);
    if (lane >= off) incl *= y;
  }
  float excl = __shfl_up(incl, 1);
  if (lane == 0) excl = 1.0f;

  float w0 = a0 * excl;
  float w1 = a1 * (excl * p1);
  float w2 = a2 * (excl * p2);
  float w3 = a3 * (excl * p3);

  // per-lane partial sums
  float dsum = w0 * tc0 + w1 * tc1 + w2 * tc2 + w3 * tc3;
  float c0 = w0 * r0.x + w1 * r0.w + w2 * r1.z + w3 * r2.y;
  float c1 = w0 * r0.y + w1 * r1.x + w2 * r1.w + w3 * r2.z;
  float c2 = w0 * r0.z + w1 * r1.y + w2 * r2.x + w3 * r2.w;

  // wave butterfly reductions
#pragma unroll
  for (int off = 16; off > 0; off >>= 1) {
    dsum += __shfl_xor(dsum, off);
    c0   += __shfl_xor(c0, off);
    c1   += __shfl_xor(c1, off);
    c2   += __shfl_xor(c2, off);
  }

  // outputs
  float4 wv = make_float4(w0, w1, w2, w3);
  *(float4*)(weights + (size_t)ray * NS + lane * 4) = wv;
  if (lane == 0) {
    depths[ray] = dsum;
    colors[3 * ray + 0] = c0;
    colors[3 * ray + 1] = c1;
    colors[3 * ray + 2] = c2;
  }
}

extern "C" void kernel_launch(void* const* d_in, const int* in_sizes, int n_in,
                              void* d_out, int out_size, void* d_ws, size_t ws_size,
                              hipStream_t stream) {
  const float* sigmas   = (const float*)d_in[0];
  const float* rgbs     = (const float*)d_in[1];
  const float* t_inters = (const float*)d_in[2];

  const int n_rays = in_sizes[0] / NS;   // 65536

  float* out     = (float*)d_out;
  float* depths  = out;                          // [N]
  float* colors  = out + n_rays;                 // [N,3]
  float* weights = out + (size_t)n_rays * 4;     // [N,128]

  dim3 grid(n_rays / RPB);
  dim3 block(RPB * 32);
  volume_render_kernel<<<grid, block, 0, stream>>>(sigmas, rgbs, t_inters,
                                                   depths, colors, weights);
}